// SimulatedNoiseRegressionEval_62801011802432
// MI455X (gfx1250) — compile-verified
//
#include <hip/hip_runtime.h>
#include <math.h>

// ---------------------------------------------------------------------------
// Periodic KNN (k=12) for 2048 atoms in a 3x3x3 supercell, MI455X / gfx1250.
//
// d2(r,j) = |c_r|^2 + |p_j|^2 - 2 c_r . p_j
// We compute s(r,j) = q_j - 2 c_r . p_j with one V_WMMA_F32_16X16X4_F32 per
// 16x16 tile:  A row = [cx,cy,cz,1], B col = [-2px,-2py,-2pz,q],
// q = |p|^2 (or 1e30 if the supercell point fails the validity test).
// s differs from d2 by the row-constant |c|^2, so per-row top-k order is
// identical; |c|^2 is added back only for the 12 selected edges.
// ---------------------------------------------------------------------------

#define N_ATOMS   2048
#define NCELLS    27
#define NSUP      (NCELLS * N_ATOMS)        // 55296
#define KSEL      12
#define SLICE     1024                      // columns staged in LDS per chunk
#define NCHUNK    (NSUP / SLICE)            // 54
#define CSPLIT    6                         // column splits (parallelism)
#define CHUNKS_PER_SPLIT (NCHUNK / CSPLIT)  // 9
#define ROWS_PER_BLOCK 16
#define TPB       256

#define OFF_DIST  (N_ATOMS * KSEL * 3)      // 73728
#define OFF_SRC   (OFF_DIST + N_ATOMS * KSEL)
#define OFF_DST   (OFF_SRC  + N_ATOMS * KSEL)
#define OFF_NUM   (OFF_DST  + N_ATOMS * KSEL)

typedef __attribute__((ext_vector_type(2))) float v2f;
typedef __attribute__((ext_vector_type(8))) float v8f;

// ---------------------------------------------------------------------------
// Kernel 1: build packed B matrix Bp[j] = (-2px, -2py, -2pz, q) in workspace.
// ---------------------------------------------------------------------------
__global__ __launch_bounds__(256)
void build_bpacked(const float* __restrict__ pos, const float* __restrict__ cell,
                   const int* __restrict__ numbers, float4* __restrict__ Bp,
                   float* __restrict__ outNum) {
  const int j = blockIdx.x * blockDim.x + threadIdx.x;
  if (j >= NSUP) return;

  const float m0 = cell[0], m1 = cell[1], m2 = cell[2];
  const float m3 = cell[3], m4 = cell[4], m5 = cell[5];
  const float m6 = cell[6], m7 = cell[7], m8 = cell[8];

  const int ci = j >> 11;               // cell replica index 0..26
  const int i  = j & (N_ATOMS - 1);     // atom index
  const float ox = (float)(ci / 9 - 1);
  const float oy = (float)((ci / 3) % 3 - 1);
  const float oz = (float)(ci % 3 - 1);

  const float px = pos[i*3+0] + ox*m0 + oy*m3 + oz*m6;
  const float py = pos[i*3+1] + ox*m1 + oy*m4 + oz*m7;
  const float pz = pos[i*3+2] + ox*m2 + oy*m5 + oz*m8;

  // cofactors of cell (row-major), frac = p @ inv(cell)
  const float c00 =  (m4*m8 - m5*m7);
  const float c01 = -(m3*m8 - m5*m6);
  const float c02 =  (m3*m7 - m4*m6);
  const float c10 = -(m1*m8 - m2*m7);
  const float c11 =  (m0*m8 - m2*m6);
  const float c12 = -(m0*m7 - m1*m6);
  const float c20 =  (m1*m5 - m2*m4);
  const float c21 = -(m0*m5 - m2*m3);
  const float c22 =  (m0*m4 - m1*m3);
  const float inv_det = 1.0f / (m0*c00 + m1*c01 + m2*c02);

  const float f0 = (px*c00 + py*c01 + pz*c02) * inv_det;
  const float f1 = (px*c10 + py*c11 + pz*c12) * inv_det;
  const float f2 = (px*c20 + py*c21 + pz*c22) * inv_det;

  const float rn0 = sqrtf(m0*m0 + m1*m1 + m2*m2);
  const float rn1 = sqrtf(m3*m3 + m4*m4 + m5*m5);
  const float rn2 = sqrtf(m6*m6 + m7*m7 + m8*m8);
  const float dx  = 0.1f * fminf(rn0, fminf(rn1, rn2));
  const float e0 = dx / rn0, e1 = dx / rn1, e2 = dx / rn2;

  const bool valid = (f0 >= -e0) & (f0 <= 1.0f + e0) &
                     (f1 >= -e1) & (f1 <= 1.0f + e1) &
                     (f2 >= -e2) & (f2 <= 1.0f + e2);

  const float q = valid ? (px*px + py*py + pz*pz) : 1.0e30f;
  Bp[j] = make_float4(-2.0f*px, -2.0f*py, -2.0f*pz, q);

  if (j < N_ATOMS) outNum[j] = (float)numbers[j];   // passthrough output
}

// ---------------------------------------------------------------------------
// Kernel 2: WMMA distance tiles + per-row partial top-12 over a column span.
// Grid: (128 row-stripes, CSPLIT column splits), 256 threads = 8 wave32s.
// ---------------------------------------------------------------------------
__global__ __launch_bounds__(TPB)
void knn_partial(const float* __restrict__ pos, const float* __restrict__ Bpf,
                 float* __restrict__ candV, int* __restrict__ candI) {
  __shared__ float stage[ROWS_PER_BLOCK * SLICE];           // 64 KB
  __shared__ float mval[ROWS_PER_BLOCK * 16 * KSEL];        // 12 KB
  __shared__ int   midx[ROWS_PER_BLOCK * 16 * KSEL];        // 12 KB

  const int tid   = threadIdx.x;
  const int lane  = tid & 31;
  const int wave  = tid >> 5;                               // 0..7
  const int row0  = blockIdx.x * ROWS_PER_BLOCK;
  const int split = blockIdx.y;
  const int chunk0 = split * CHUNKS_PER_SPLIT;

  // A fragment (16x4 f32): lanes 0-15 hold K={0,1}, lanes 16-31 hold K={2,3}
  const int   am = lane & 15;
  const float ax = pos[(row0 + am)*3 + 0];
  const float ay = pos[(row0 + am)*3 + 1];
  const float az = pos[(row0 + am)*3 + 2];
  v2f a;
  a.x = (lane < 16) ? ax : az;     // K=0 : K=2
  a.y = (lane < 16) ? ay : 1.0f;   // K=1 : K=3  (pad carries the q term via B)

  // top-k scan assignment: 16 lanes per row
  const int trow  = tid >> 4;           // local row 0..15
  const int tsub  = tid & 15;
  const int grow  = row0 + trow;
  const int selfj = 13 * N_ATOMS + grow;

  float tv[KSEL]; int tix[KSEL];
#pragma unroll
  for (int u = 0; u < KSEL; ++u) { tv[u] = 3.0e38f; tix[u] = 0; }

  for (int ch = 0; ch < CHUNKS_PER_SPLIT; ++ch) {
    const int chunk = chunk0 + ch;
    const int cbase = chunk * SLICE;

    // --- fill 16 x SLICE slab of s-values with WMMA (8 tiles per wave) ---
    for (int t8 = 0; t8 < 8; ++t8) {
      const int tile = wave * 8 + t8;                       // 0..63
      const int j    = cbase + tile * 16 + (lane & 15);
      // B fragment: one 8-byte load per lane (K-pair selected by lane half)
      const v2f b = *(const v2f*)(Bpf + (size_t)j * 4 + ((lane >> 4) << 1));
      v8f acc = {};
      acc = __builtin_amdgcn_wmma_f32_16x16x4_f32(
          /*neg_a=*/false, a, /*neg_b=*/false, b,
          /*c_mod=*/(short)0, acc, /*reuse_a=*/false, /*reuse_b=*/false);
      const int n     = lane & 15;
      const int mbase = (lane >> 4) * 8;
#pragma unroll
      for (int v = 0; v < 8; ++v)
        stage[(mbase + v) * SLICE + tile * 16 + n] = acc[v];
    }
    __syncthreads();

    // prefetch next chunk's B data while we scan this one
    if (ch + 1 < CHUNKS_PER_SPLIT && tid < 128)
      __builtin_prefetch(Bpf + (size_t)(cbase + SLICE) * 4 + tid * 32, 0, 1);

    // --- register-resident top-12 insertion (fully unrolled, no spills) ---
    for (int cc = tsub; cc < SLICE; cc += 16) {
      const float s = stage[trow * SLICE + cc];
      const int   j = cbase + cc;
      if (j != selfj && s < tv[KSEL - 1]) {
        float cv = s; int cj = j;
#pragma unroll
        for (int u = 0; u < KSEL; ++u) {
          if (cv < tv[u]) {
            const float tf = tv[u]; const int tt = tix[u];
            tv[u] = cv; tix[u] = cj; cv = tf; cj = tt;
          }
        }
      }
    }
    __syncthreads();
  }

  // --- merge the 16 per-lane lists of each row -> 12 candidates/row/split ---
#pragma unroll
  for (int u = 0; u < KSEL; ++u) {
    mval[(trow * 16 + tsub) * KSEL + u] = tv[u];
    midx[(trow * 16 + tsub) * KSEL + u] = tix[u];
  }
  __syncthreads();

  if (tid < ROWS_PER_BLOCK) {
    float* base = &mval[tid * 16 * KSEL];
    int*   ib   = &midx[tid * 16 * KSEL];
    const int outBase = ((row0 + tid) * CSPLIT + split) * KSEL;
    for (int sel = 0; sel < KSEL; ++sel) {
      float best = 3.5e38f; int bp = 0;
      for (int q = 0; q < 16 * KSEL; ++q) {
        const float v = base[q];
        if (v < best) { best = v; bp = q; }
      }
      candV[outBase + sel] = best;
      candI[outBase + sel] = ib[bp];
      base[bp] = 3.6e38f;     // mark consumed
    }
  }
}

// ---------------------------------------------------------------------------
// Kernel 3: merge CSPLIT*12 candidates per row, emit edges in ascending-d2
// order (matches top_k of -d2). One thread per atom.
// ---------------------------------------------------------------------------
__global__ __launch_bounds__(128)
void knn_finalize(const float* __restrict__ pos, const float4* __restrict__ Bp,
                  const float* __restrict__ candV, const int* __restrict__ candI,
                  float* __restrict__ out) {
  constexpr int M = CSPLIT * KSEL;                          // 72
  __shared__ float lv[128 * M];                             // 36 KB
  __shared__ int   li[128 * M];                             // 36 KB
  const int r = blockIdx.x * 128 + threadIdx.x;             // atom 0..2047
  const int t = threadIdx.x;

  for (int u = 0; u < M; ++u) {
    lv[t * M + u] = candV[r * M + u];
    li[t * M + u] = candI[r * M + u];
  }

  const float cx = pos[r*3+0], cy = pos[r*3+1], cz = pos[r*3+2];
  const float c2 = cx*cx + cy*cy + cz*cz;

  for (int sel = 0; sel < KSEL; ++sel) {
    float best = 3.5e38f; int bp = 0;
    for (int q = 0; q < M; ++q) {
      const float v = lv[t * M + q];
      if (v < best) { best = v; bp = q; }
    }
    lv[t * M + bp] = 3.6e38f;
    const int j = li[t * M + bp];

    const float4 bb = Bp[j];
    const float px = -0.5f * bb.x, py = -0.5f * bb.y, pz = -0.5f * bb.z;
    float d2 = best + c2; d2 = d2 > 0.0f ? d2 : 0.0f;

    const int e = r * KSEL + sel;
    out[e*3 + 0] = px - cx;                 // edge_vec
    out[e*3 + 1] = py - cy;
    out[e*3 + 2] = pz - cz;
    out[OFF_DIST + e] = sqrtf(d2);          // dist
    out[OFF_SRC  + e] = (float)(j & (N_ATOMS - 1));  // src_atom (j % N)
    out[OFF_DST  + e] = (float)r;           // dst_atom
  }
}

// ---------------------------------------------------------------------------
extern "C" void kernel_launch(void* const* d_in, const int* in_sizes, int n_in,
                              void* d_out, int out_size, void* d_ws, size_t ws_size,
                              hipStream_t stream) {
  (void)in_sizes; (void)n_in; (void)out_size; (void)ws_size;
  const float* pos     = (const float*)d_in[0];   // 2048 x 3
  const float* cell    = (const float*)d_in[1];   // 3 x 3
  const int*   numbers = (const int*)d_in[2];     // 2048
  // d_in[3] is k == 12 (compile-time here)
  float* out = (float*)d_out;

  // workspace layout: Bp | candV | candI  (~2.06 MB total)
  char*   ws    = (char*)d_ws;
  float4* Bp    = (float4*)ws;
  float*  candV = (float*)(ws + (size_t)NSUP * sizeof(float4));
  int*    candI = (int*)  (ws + (size_t)NSUP * sizeof(float4)
                              + (size_t)N_ATOMS * CSPLIT * KSEL * sizeof(float));

  build_bpacked<<<(NSUP + 255) / 256, 256, 0, stream>>>(pos, cell, numbers, Bp,
                                                        out + OFF_NUM);
  dim3 gridB(N_ATOMS / ROWS_PER_BLOCK, CSPLIT);             // 128 x 6
  knn_partial<<<gridB, TPB, 0, stream>>>(pos, (const float*)Bp, candV, candI);
  knn_finalize<<<N_ATOMS / 128, 128, 0, stream>>>(pos, Bp, candV, candI, out);
}